// NodeBlock_90623809945606
// MI455X (gfx1250) — compile-verified
//
#include <hip/hip_runtime.h>

typedef float v2f __attribute__((ext_vector_type(2)));
typedef float v8f __attribute__((ext_vector_type(8)));

#define N_NODES 100000
#define NUM_E   1200000
#define DE 64
#define DV 64
#define HD 128
#define DIN 128   // DV + DE

// ---------------------------------------------------------------------------
// Kernel 1: zero the scatter accumulation buffer (v1) in workspace.
// 100000*64 floats = 1.6M float4 stores, grid-stride.
// ---------------------------------------------------------------------------
__global__ void zero_ws(float4* __restrict__ v1f4, int total_f4) {
    int i = blockIdx.x * blockDim.x + threadIdx.x;
    const float4 z = {0.f, 0.f, 0.f, 0.f};
    for (; i < total_f4; i += gridDim.x * blockDim.x)
        v1f4[i] = z;
}

// ---------------------------------------------------------------------------
// Kernel 2: double scatter-add of edge features into v1 at both endpoints.
// One thread per (edge, 4-feature chunk): 1x b128 load + 8x atomic_add_f32.
// v1 (25.6MB) lives in L2 (192MB) so the atomic RMW traffic stays on-chip.
// ---------------------------------------------------------------------------
__global__ void scatter_edges(const float* __restrict__ e,
                              const int*   __restrict__ edges,
                              float*       __restrict__ v1) {
    long long idx = (long long)blockIdx.x * blockDim.x + threadIdx.x;
    const long long total = (long long)NUM_E * (DE / 4);
    if (idx >= total) return;
    const int edge = (int)(idx >> 4);        // DE/4 == 16 chunks per edge
    const int c    = (int)(idx & 15) * 4;

    const float4 ev = *(const float4*)(e + (long long)edge * DE + c);
    const int s = edges[2 * edge + 0];
    const int d = edges[2 * edge + 1];

    float* ps = v1 + (long long)s * DE + c;
    float* pd = v1 + (long long)d * DE + c;
    atomicAdd(ps + 0, ev.x); atomicAdd(ps + 1, ev.y);
    atomicAdd(ps + 2, ev.z); atomicAdd(ps + 3, ev.w);
    atomicAdd(pd + 0, ev.x); atomicAdd(pd + 1, ev.y);
    atomicAdd(pd + 2, ev.z); atomicAdd(pd + 3, ev.w);
}

// ---------------------------------------------------------------------------
// Kernel 3: fused per-node MLP using native f32 WMMA (16x16x4).
// One wave32 per 16-node tile. LDS stages the 16x128 input tile; it is
// overwritten with the 16x128 hidden tile (post bias+ReLU) between GEMMs.
//
// Fragment layouts per CDNA5 ISA 7.12.2:
//   A (16x4 f32): lane L<16 holds row M=L, {K,K+1}; lane L+16 holds {K+2,K+3}
//   B (4x16 f32): lane L<16 holds col N=L, {K,K+1}; lane L+16 holds {K+2,K+3}
//   C/D (16x16):  element i of lane L = D[(L>>4)*8 + i][L & 15]
// ---------------------------------------------------------------------------
__global__ void __launch_bounds__(32)
node_mlp(const float* __restrict__ v,    // [N, DV]
         const float* __restrict__ v1,   // [N, DE]  (workspace)
         const float* __restrict__ W1,   // [DIN, HD]
         const float* __restrict__ b1,   // [HD]
         const float* __restrict__ W2,   // [HD, DV]
         const float* __restrict__ b2,   // [DV]
         float*       __restrict__ out)  // [N, DV]
{
    __shared__ float tile[16][DIN];      // 8 KB: v2 tile, then H tile

    const int lane  = threadIdx.x;       // wave32
    const int node0 = blockIdx.x * 16;

    // --- Stage v2 = [v | v1] tile into LDS (coalesced float4) ---------------
    #pragma unroll
    for (int j = 0; j < 16; ++j) {
        const int f4    = lane + 32 * j;     // 512 float4 total
        const int row   = f4 >> 5;           // 32 float4 per row
        const int col   = (f4 & 31) * 4;
        float4 val;
        if (col < DV)
            val = *(const float4*)(v  + (long long)(node0 + row) * DV + col);
        else
            val = *(const float4*)(v1 + (long long)(node0 + row) * DE + (col - DV));
        *(float4*)&tile[row][col] = val;
    }
    __syncthreads();

    const int m    = lane & 15;          // A row / B col index within tile
    const int koff = (lane >> 4) * 2;    // K sub-offset per lane half
    const int mrow = (lane >> 4) * 8;    // C/D row base
    const int ncol = lane & 15;          // C/D col

    // --- GEMM1: H = relu(v2 @ W1 + b1), 8 column tiles of 16 ----------------
    v8f acc[8] = {};
    for (int k = 0; k < DIN; k += 4) {
        v2f a;
        a.x = tile[m][k + koff];
        a.y = tile[m][k + koff + 1];
        #pragma unroll
        for (int t = 0; t < 8; ++t) {
            const int col = t * 16 + ncol;
            v2f b;
            b.x = W1[(k + koff)     * HD + col];
            b.y = W1[(k + koff + 1) * HD + col];
            acc[t] = __builtin_amdgcn_wmma_f32_16x16x4_f32(
                false, a, false, b, (short)0, acc[t], false, false);
        }
    }
    __syncthreads();

    // --- bias + ReLU, store H back into LDS tile ----------------------------
    #pragma unroll
    for (int t = 0; t < 8; ++t) {
        const float bias = b1[t * 16 + ncol];
        #pragma unroll
        for (int i = 0; i < 8; ++i) {
            const float h = acc[t][i] + bias;
            tile[mrow + i][t * 16 + ncol] = fmaxf(h, 0.f);
        }
    }
    __syncthreads();

    // --- GEMM2: out = H @ W2 + b2, 4 column tiles of 16 ---------------------
    v8f acc2[4] = {};
    for (int k = 0; k < HD; k += 4) {
        v2f a;
        a.x = tile[m][k + koff];
        a.y = tile[m][k + koff + 1];
        #pragma unroll
        for (int t = 0; t < 4; ++t) {
            const int col = t * 16 + ncol;
            v2f b;
            b.x = W2[(k + koff)     * DV + col];
            b.y = W2[(k + koff + 1) * DV + col];
            acc2[t] = __builtin_amdgcn_wmma_f32_16x16x4_f32(
                false, a, false, b, (short)0, acc2[t], false, false);
        }
    }

    // --- bias + coalesced store --------------------------------------------
    #pragma unroll
    for (int t = 0; t < 4; ++t) {
        const float bias = b2[t * 16 + ncol];
        #pragma unroll
        for (int i = 0; i < 8; ++i) {
            out[(long long)(node0 + mrow + i) * DV + t * 16 + ncol] =
                acc2[t][i] + bias;
        }
    }
}

// ---------------------------------------------------------------------------
extern "C" void kernel_launch(void* const* d_in, const int* in_sizes, int n_in,
                              void* d_out, int out_size, void* d_ws, size_t ws_size,
                              hipStream_t stream) {
    const float* e     = (const float*)d_in[0];
    const float* v     = (const float*)d_in[1];
    const int*   edges = (const int*)  d_in[2];
    const float* W1    = (const float*)d_in[3];
    const float* b1    = (const float*)d_in[4];
    const float* W2    = (const float*)d_in[5];
    const float* b2    = (const float*)d_in[6];
    float* out = (float*)d_out;
    float* v1  = (float*)d_ws;           // [N, DE] scatter accumulator

    // 1) zero the accumulator (deterministic per call)
    const int total_f4 = N_NODES * DE / 4;           // 1.6M
    zero_ws<<<4096, 256, 0, stream>>>((float4*)v1, total_f4);

    // 2) double scatter-add
    const long long sc_total = (long long)NUM_E * (DE / 4);
    const int sthreads = 256;
    const int sblocks  = (int)((sc_total + sthreads - 1) / sthreads);
    scatter_edges<<<sblocks, sthreads, 0, stream>>>(e, edges, v1);

    // 3) fused MLP via f32 WMMA, one wave32 per 16-node tile
    node_mlp<<<N_NODES / 16, 32, 0, stream>>>(v, v1, W1, b1, W2, b2, out);
}